// Model_670014898178
// MI455X (gfx1250) — compile-verified
//
#include <hip/hip_runtime.h>
#include <math.h>

// ---------------------------------------------------------------------------
// MI455X (gfx1250) implementation.
//
// Dominant cost: fcc GEMM reads 134MB weights + 67MB activations ~= 9us at
// 23.3 TB/s HBM. conv2 (4.8 GFLOP f32) is routed through V_WMMA_F32_16X16X4_F32
// so it doesn't dominate on the VALU. All matrix math stays f32 (K=262144
// reduction; bf16 inputs would risk accuracy) -> v_wmma_f32_16x16x4_f32.
// Split-K partials are reduced in a fixed order (no float atomics) so the
// whole pipeline is bitwise deterministic across graph replays.
// ---------------------------------------------------------------------------

typedef __attribute__((ext_vector_type(2))) float v2f;
typedef __attribute__((ext_vector_type(8))) float v8f;

#define K_FCC 262144            // 16*128*128
#define SLABS 128
#define KSLAB (K_FCC / SLABS)   // 2048

// D(16x16 f32) = A(16x4 f32) * B(4x16 f32) + C
// A layout: lanes 0-15 hold M=lane, K={k0,k0+1}; lanes 16-31 M=lane-16, K={k0+2,k0+3}
// B layout: lanes 0-15 hold N=lane, K rows {k0,k0+1}; lanes 16-31 K rows {k0+2,k0+3}
// C/D layout: vgpr i, lane l: M = i + 8*(l>=16), N = l&15
__device__ __forceinline__ v8f wmma4(v2f a, v2f b, v8f c) {
  return __builtin_amdgcn_wmma_f32_16x16x4_f32(
      /*neg_a=*/false, a, /*neg_b=*/false, b,
      /*c_mod=*/(short)0, c, /*reuse_a=*/false, /*reuse_b=*/false);
}

// ---------------------------------------------------------------------------
// GCN path (tiny). Rows 0..127 of the (8192,5) input are batch 0; the
// reference's mean-replace makes layer-1 rows 0..127 identical, so layer-2's
// mean-replace is an identity. We only need xbar = mean over x1[0] rows.
// ---------------------------------------------------------------------------
__global__ __launch_bounds__(128) void gcn_xbar_kernel(
    const float* __restrict__ x1, float* __restrict__ xbar) {
  __shared__ float s[128 * 5];
  const int t = threadIdx.x;           // 128 threads
  #pragma unroll
  for (int f = 0; f < 5; ++f) s[t * 5 + f] = x1[t * 5 + f];
  __syncthreads();
  if (t < 5) {
    float acc = 0.f;
    for (int n = 0; n < 128; ++n) acc += s[n * 5 + t];
    xbar[t] = acc * (1.f / 128.f);
  }
}

__global__ __launch_bounds__(256) void gcn_h_kernel(
    const float* __restrict__ x1, const float* __restrict__ xbar,
    const float* __restrict__ W1, const float* __restrict__ b1,
    const float* __restrict__ W2, const float* __restrict__ b2,
    float* __restrict__ h_buf) {
  const int r = blockIdx.x * 256 + threadIdx.x;   // 0..8191
  float xr[5];
  if (r < 128) {
    #pragma unroll
    for (int f = 0; f < 5; ++f) xr[f] = xbar[f];
  } else {
    #pragma unroll
    for (int f = 0; f < 5; ++f) xr[f] = x1[r * 5 + f];
  }
  float acc2 = b2[0];
  #pragma unroll
  for (int j = 0; j < 10; ++j) {
    float t = b1[j];
    #pragma unroll
    for (int f = 0; f < 5; ++f) t += xr[f] * W1[f * 10 + j];
    t = t > 0.f ? t : 0.f;            // relu after gcn1
    acc2 += t * W2[j];                // gcn2 (10 -> 1)
  }
  h_buf[r] = acc2 > 0.f ? acc2 : 0.f; // relu after gcn2
}

__global__ __launch_bounds__(128) void fcg_kernel(
    const float* __restrict__ h_buf, const float* __restrict__ W,
    const float* __restrict__ bias, float* __restrict__ g_buf) {
  __shared__ float sh[128];
  const int b = blockIdx.x, j = threadIdx.x;
  sh[j] = h_buf[b * 128 + j];
  __syncthreads();
  float acc = bias[j];
  for (int i = 0; i < 128; ++i) acc += sh[i] * W[i * 128 + j];
  g_buf[b * 128 + j] = acc > 0.f ? acc : 0.f;
}

// ---------------------------------------------------------------------------
// Fused conv1+conv2 per (batch, 16x16 tile). conv1 -> LDS (18x18x16 with
// conv2 halo), conv2 as implicit GEMM  W2(16x144) @ im2col(144x256) using
// WMMA f32 16x16x4. Output stored in the reference reshape layout:
// c[b][ch*16384 + r*128 + c].
// ---------------------------------------------------------------------------
__global__ __launch_bounds__(256) void conv_fused_kernel(
    const float* __restrict__ x2,
    const float* __restrict__ w1, const float* __restrict__ b1,
    const float* __restrict__ w2, const float* __restrict__ b2,
    float* __restrict__ c_buf) {
  __shared__ float s_in[20 * 20];      // input tile + halo 2 (zero padded)
  __shared__ float s_w1[16 * 9];
  __shared__ float s_b1[16];
  __shared__ float s_w2[16 * 144];     // A matrix: [ch_out][ch_in*9 + ky*3 + kx]
  __shared__ float s_b2[16];
  __shared__ float s_c1[16 * 324];     // conv1 out: [ch_in][18*18], relu'd

  const int tid = threadIdx.x;
  const int bid = blockIdx.x;
  const int b  = bid >> 6;
  const int tr = (bid >> 3) & 7;
  const int tc = bid & 7;
  const int r0 = tr * 16, c0 = tc * 16;

  for (int idx = tid; idx < 400; idx += 256) {
    const int ri = r0 - 2 + idx / 20;
    const int ci = c0 - 2 + idx % 20;
    float v = 0.f;
    if (ri >= 0 && ri < 128 && ci >= 0 && ci < 128)
      v = x2[(size_t)b * 16384 + ri * 128 + ci];
    s_in[idx] = v;
  }
  for (int idx = tid; idx < 144; idx += 256) s_w1[idx] = w1[idx];
  for (int idx = tid; idx < 2304; idx += 256) s_w2[idx] = w2[idx];
  if (tid < 16) { s_b1[tid] = b1[tid]; s_b2[tid] = b2[tid]; }
  __syncthreads();

  // conv1 (+bias, relu) on the 18x18 halo region; positions outside the
  // 128x128 image are conv2's zero padding -> store exact 0.
  for (int idx = tid; idx < 16 * 324; idx += 256) {
    const int ch = idx / 324;
    const int p  = idx - ch * 324;
    const int pr = p / 18, pc = p - pr * 18;
    const int gr = r0 - 1 + pr, gc = c0 - 1 + pc;
    float v = 0.f;
    if (gr >= 0 && gr < 128 && gc >= 0 && gc < 128) {
      float acc = s_b1[ch];
      const float* wr = &s_w1[ch * 9];
      #pragma unroll
      for (int ky = 0; ky < 3; ++ky)
        #pragma unroll
        for (int kx = 0; kx < 3; ++kx)
          acc += s_in[(pr + ky) * 20 + pc + kx] * wr[ky * 3 + kx];
      v = acc > 0.f ? acc : 0.f;
    }
    s_c1[idx] = v;
  }
  __syncthreads();

  // conv2 via WMMA: each wave owns 2 output rows (N = 16 columns each).
  const int lane = tid & 31;
  const int wave = tid >> 5;
  const int half = lane >> 4;          // 0: K={k0,k0+1}, 1: K={k0+2,k0+3}
  const int nn   = lane & 15;          // M index for A, N index for B/D

  v8f acc0 = {0, 0, 0, 0, 0, 0, 0, 0};
  v8f acc1 = acc0;
  const int rA = 2 * wave;
  const int rB = 2 * wave + 1;

  for (int k0 = 0; k0 < 144; k0 += 4) {
    const int ka = k0 + 2 * half;
    v2f a = *(const v2f*)&s_w2[nn * 144 + ka];   // 8B-aligned ds_load_b64
    const int ch0 = ka / 9,      t0 = ka - ch0 * 9;
    const int dy0 = t0 / 3,      dx0 = t0 - dy0 * 3;
    const int k1  = ka + 1;
    const int ch1 = k1 / 9,      t1 = k1 - ch1 * 9;
    const int dy1 = t1 / 3,      dx1 = t1 - dy1 * 3;
    v2f bA, bB;
    bA.x = s_c1[ch0 * 324 + (rA + dy0) * 18 + nn + dx0];
    bA.y = s_c1[ch1 * 324 + (rA + dy1) * 18 + nn + dx1];
    bB.x = s_c1[ch0 * 324 + (rB + dy0) * 18 + nn + dx0];
    bB.y = s_c1[ch1 * 324 + (rB + dy1) * 18 + nn + dx1];
    acc0 = wmma4(a, bA, acc0);
    acc1 = wmma4(a, bB, acc1);
  }

  #pragma unroll
  for (int i = 0; i < 8; ++i) {
    const int ch = i + 8 * half;
    const float bias = s_b2[ch];
    float v0 = acc0[i] + bias; v0 = v0 > 0.f ? v0 : 0.f;
    float v1 = acc1[i] + bias; v1 = v1 > 0.f ? v1 : 0.f;
    const size_t base = (size_t)b * 262144 + (size_t)ch * 16384 + (size_t)c0 + nn;
    c_buf[base + (size_t)(r0 + rA) * 128] = v0;
    c_buf[base + (size_t)(r0 + rB) * 128] = v1;
  }
}

// ---------------------------------------------------------------------------
// fcc GEMM: (64 x 262144) @ (262144 x 128), split-K into 128 slabs.
// Block = (slab, mtile); all 8 waves share the same 16xKSLAB A slab (L0/L2
// reuse), wave w computes N-tile w. Partials written per-slab (deterministic).
// ---------------------------------------------------------------------------
__global__ __launch_bounds__(256) void fcc_gemm_kernel(
    const float* __restrict__ A, const float* __restrict__ Bw,
    float* __restrict__ partial) {
  const int slab  = blockIdx.x >> 2;
  const int mtile = blockIdx.x & 3;
  const int wave  = threadIdx.x >> 5;
  const int lane  = threadIdx.x & 31;
  const int half  = lane >> 4;
  const int nn    = lane & 15;
  const int m0    = mtile * 16;
  const int n0    = wave * 16;
  const size_t k0 = (size_t)slab * KSLAB;

  const float* arow = A + (size_t)(m0 + nn) * K_FCC + k0 + 2 * half;
  const float* bcol = Bw + (k0 + 2 * half) * 128 + n0 + nn;

  v8f acc = {0, 0, 0, 0, 0, 0, 0, 0};
  #pragma unroll 4
  for (int k = 0; k < KSLAB; k += 4) {
    v2f a = *(const v2f*)(arow + k);        // contiguous 8B per lane
    v2f bb;
    bb.x = bcol[(size_t)k * 128];           // row k   of B, col n0+nn
    bb.y = bcol[(size_t)k * 128 + 128];     // row k+1 of B
    acc = wmma4(a, bb, acc);
  }

  float* out = partial + (size_t)slab * 8192 + (size_t)n0 + nn;
  #pragma unroll
  for (int i = 0; i < 8; ++i)
    out[(size_t)(m0 + i + 8 * half) * 128] = acc[i];
}

// ---------------------------------------------------------------------------
// Head: fixed-order slab reduction + fcc bias/relu, concat [c, g],
// fc1(256->128) relu, fc2(128->64) relu, fc3(64->2) sigmoid.
// ---------------------------------------------------------------------------
__global__ __launch_bounds__(128) void head_kernel(
    const float* __restrict__ partial, const float* __restrict__ fcc_b,
    const float* __restrict__ g_buf,
    const float* __restrict__ fc1_w, const float* __restrict__ fc1_b,
    const float* __restrict__ fc2_w, const float* __restrict__ fc2_b,
    const float* __restrict__ fc3_w, const float* __restrict__ fc3_b,
    float* __restrict__ out) {
  __shared__ float sx[256];
  __shared__ float sh1[128];
  __shared__ float sh2[64];
  const int b = blockIdx.x, j = threadIdx.x;

  float acc = 0.f;
  for (int s = 0; s < SLABS; ++s) acc += partial[(size_t)s * 8192 + b * 128 + j];
  acc += fcc_b[j];
  sx[j]       = acc > 0.f ? acc : 0.f;   // c half
  sx[128 + j] = g_buf[b * 128 + j];      // g half
  __syncthreads();

  float a1 = fc1_b[j];
  for (int i = 0; i < 256; ++i) a1 += sx[i] * fc1_w[i * 128 + j];
  sh1[j] = a1 > 0.f ? a1 : 0.f;
  __syncthreads();

  if (j < 64) {
    float a2 = fc2_b[j];
    for (int i = 0; i < 128; ++i) a2 += sh1[i] * fc2_w[i * 64 + j];
    sh2[j] = a2 > 0.f ? a2 : 0.f;
  }
  __syncthreads();

  if (j < 2) {
    float a3 = fc3_b[j];
    for (int i = 0; i < 64; ++i) a3 += sh2[i] * fc3_w[i * 2 + j];
    out[b * 2 + j] = 1.f / (1.f + expf(-a3));
  }
}

extern "C" void kernel_launch(void* const* d_in, const int* in_sizes, int n_in,
                              void* d_out, int out_size, void* d_ws, size_t ws_size,
                              hipStream_t stream) {
  const float* x1      = (const float*)d_in[0];
  const float* x2      = (const float*)d_in[1];
  const float* gcn1_w  = (const float*)d_in[2];
  const float* gcn1_b  = (const float*)d_in[3];
  const float* gcn2_w  = (const float*)d_in[4];
  const float* gcn2_b  = (const float*)d_in[5];
  const float* fcg_w   = (const float*)d_in[6];
  const float* fcg_b   = (const float*)d_in[7];
  const float* conv1_w = (const float*)d_in[8];
  const float* conv1_b = (const float*)d_in[9];
  const float* conv2_w = (const float*)d_in[10];
  const float* conv2_b = (const float*)d_in[11];
  const float* fcc_w   = (const float*)d_in[12];
  const float* fcc_b   = (const float*)d_in[13];
  const float* fc1_w   = (const float*)d_in[14];
  const float* fc1_b   = (const float*)d_in[15];
  const float* fc2_w   = (const float*)d_in[16];
  const float* fc2_b   = (const float*)d_in[17];
  const float* fc3_w   = (const float*)d_in[18];
  const float* fc3_b   = (const float*)d_in[19];
  float* out = (float*)d_out;

  // workspace layout (floats); every region is fully written before read
  float* ws     = (float*)d_ws;
  float* c_buf  = ws;                                   // 16,777,216 (64x262144)
  float* part   = c_buf + (size_t)16777216;             // SLABS*8192
  float* h_buf  = part + (size_t)SLABS * 8192;          // 8192
  float* g_buf  = h_buf + 8192;                         // 8192
  float* xbar   = g_buf + 8192;                         // 5

  gcn_xbar_kernel<<<1, 128, 0, stream>>>(x1, xbar);
  gcn_h_kernel<<<32, 256, 0, stream>>>(x1, xbar, gcn1_w, gcn1_b, gcn2_w, gcn2_b, h_buf);
  fcg_kernel<<<64, 128, 0, stream>>>(h_buf, fcg_w, fcg_b, g_buf);
  conv_fused_kernel<<<64 * 64, 256, 0, stream>>>(x2, conv1_w, conv1_b, conv2_w, conv2_b, c_buf);
  fcc_gemm_kernel<<<SLABS * 4, 256, 0, stream>>>(c_buf, fcc_w, part);
  head_kernel<<<64, 128, 0, stream>>>(part, fcc_b, g_buf,
                                      fc1_w, fc1_b, fc2_w, fc2_b, fc3_w, fc3_b, out);
}